// DualStreamBiMambaBackbone_54520314856007
// MI455X (gfx1250) — compile-verified
//
#include <hip/hip_runtime.h>
#include <cstdint>
#include <cstddef>

// ---------------- problem constants ----------------
#define B_N     64
#define L_N     2048
#define DM      32      // d_model per stream (D_LEN == D_STRUCT == 32)
#define DI      64      // d_inner = EXPAND * DM
#define NS      4       // streams: len_fwd, len_bwd, struct_fwd, struct_bwd
#define NSTATE  16      // D_STATE
#define PRW     48      // padded width of x-proj output (34 -> 48)

typedef float v2f __attribute__((ext_vector_type(2)));
typedef float v8f __attribute__((ext_vector_type(8)));

struct Ptr4 { const float* p[4]; };

__device__ __forceinline__ float sigmoidf_(float x) { return 1.f / (1.f + __expf(-x)); }
__device__ __forceinline__ float siluf_(float x)    { return x * sigmoidf_(x); }
__device__ __forceinline__ float geluf_(float x)    { return 0.5f * x * (1.f + erff(x * 0.70710678118654752f)); }

// ---------------- embedding: x -> 4 stream inputs (fwd + pre-flipped bwd) ----------------
__global__ void k_embed(const float* __restrict__ x,
                        const float* __restrict__ leW, const float* __restrict__ leB,
                        const float* __restrict__ leS, const float* __restrict__ leBt,
                        const float* __restrict__ scW, const float* __restrict__ scB,
                        const float* __restrict__ scS, const float* __restrict__ scBt,
                        const float* __restrict__ dirE,
                        float* __restrict__ X)
{
    uint32_t t = blockIdx.x * blockDim.x + threadIdx.x;
    if (t >= B_N * L_N) return;
    uint32_t b = t / L_N, l = t % L_N;
    float x0 = x[3 * t + 0], x1 = x[3 * t + 1], x2 = x[3 * t + 2];

    float pe[DM];
#pragma unroll
    for (int i = 0; i < DM / 2; ++i) {
        float fr  = __expf(-(float)(2 * i) * 0.28782313662425572f); // ln(1e4)/32
        float ang = (float)l * fr;
        pe[2 * i]     = sinf(ang);
        pe[2 * i + 1] = cosf(ang);
    }

    float v[DM];
    // ----- len stream -----
    float mu = 0.f;
#pragma unroll
    for (int d = 0; d < DM; ++d) { v[d] = x0 * leW[d] + leB[d]; mu += v[d]; }
    mu *= (1.f / DM);
    float var = 0.f;
#pragma unroll
    for (int d = 0; d < DM; ++d) { float c = v[d] - mu; var += c * c; }
    var *= (1.f / DM);
    float rs = rsqrtf(var + 1e-5f);
    uint32_t r0 = ((0u * B_N + b) * L_N + l) * DM;
    uint32_t r1 = ((1u * B_N + b) * L_N + (L_N - 1 - l)) * DM;
#pragma unroll
    for (int d = 0; d < DM; ++d) {
        float h = (v[d] - mu) * rs * leS[d] + leBt[d];
        float o = geluf_(h) + pe[d];
        X[r0 + d] = o; X[r1 + d] = o;
    }
    // ----- struct stream -----
    int di = (x1 > 0.f) ? 1 : 0;
    mu = 0.f;
#pragma unroll
    for (int d = 0; d < DM; ++d) { v[d] = x2 * scW[d] + scB[d]; mu += v[d]; }
    mu *= (1.f / DM);
    var = 0.f;
#pragma unroll
    for (int d = 0; d < DM; ++d) { float c = v[d] - mu; var += c * c; }
    var *= (1.f / DM);
    rs = rsqrtf(var + 1e-5f);
    uint32_t r2 = ((2u * B_N + b) * L_N + l) * DM;
    uint32_t r3 = ((3u * B_N + b) * L_N + (L_N - 1 - l)) * DM;
#pragma unroll
    for (int d = 0; d < DM; ++d) {
        float h = (v[d] - mu) * rs * scS[d] + scBt[d];
        float o = geluf_(h) + dirE[di * DM + d] + pe[d];
        X[r2 + d] = o; X[r3 + d] = o;
    }
}

// ---------------- layernorm over DM=32 channels, per stream params ----------------
__global__ void k_ln(const float* __restrict__ X, Ptr4 gam, Ptr4 bet, float* __restrict__ H)
{
    uint32_t t = blockIdx.x * blockDim.x + threadIdx.x;
    if (t >= NS * B_N * L_N) return;
    uint32_t s = t >> 17;                     // B_N*L_N = 2^17
    const float* g  = gam.p[s];
    const float* be = bet.p[s];
    const float* xr = X + t * DM;
    float v[DM]; float mu = 0.f;
#pragma unroll
    for (int d = 0; d < DM; ++d) { v[d] = xr[d]; mu += v[d]; }
    mu *= (1.f / DM);
    float var = 0.f;
#pragma unroll
    for (int d = 0; d < DM; ++d) { float c = v[d] - mu; var += c * c; }
    var *= (1.f / DM);
    float rs = rsqrtf(var + 1e-5f);
    float* hr = H + t * DM;
#pragma unroll
    for (int d = 0; d < DM; ++d) hr[d] = (v[d] - mu) * rs * g[d] + be[d];
}

// ---------------- generic WMMA f32 GEMM ----------------
// C = A(M,KK) @ W(Kw,Nw zero-padded to KK,N) [+bias][+resid]
// One wave = one 16x16 C tile via V_WMMA_F32_16X16X4_F32, KK/4 fully unrolled.
// W staged TRANSPOSED in LDS -> B operand pair is one ds_load_b64; A operand
// is one global_load_b64. All offsets are uint32 (largest buffer < 2^31 elems)
// so addressing stays in 32-bit VALU. grid.y = stream index.
template<int KK>
__global__ void k_gemm_wmma(const float* __restrict__ A, int lda,
                            Ptr4 W, int Kw, int Nw,
                            Ptr4 bias,
                            const float* __restrict__ resid, int ldr,
                            float* __restrict__ C, int ldc, int coff,
                            int Mrows, int N)
{
    __shared__ float sWT[64 * 128];       // transposed: sWT[n*KK + k]
    uint32_t s = blockIdx.y;
    const float* Wg = W.p[s];
    for (uint32_t i = threadIdx.x; i < (uint32_t)(KK * N); i += blockDim.x) {
        uint32_t n = i / KK, k = i % KK;
        sWT[i] = ((int)k < Kw && (int)n < Nw) ? Wg[k * Nw + n] : 0.f;
    }
    __syncthreads();

    uint32_t wid  = (blockIdx.x * blockDim.x + threadIdx.x) >> 5;
    uint32_t lane = threadIdx.x & 31;
    uint32_t tilesN = (uint32_t)N >> 4;
    uint32_t tiles  = ((uint32_t)Mrows >> 4) * tilesN;
    if (wid >= tiles) return;
    uint32_t tm = wid / tilesN, tn = wid % tilesN;
    uint32_t half = lane >> 4, l16 = lane & 15;

    uint32_t rowBase = s * (uint32_t)Mrows;
    const float* Ar = A + (rowBase + tm * 16 + l16) * (uint32_t)lda;
    uint32_t nC = tn * 16 + l16;
    const float* Bc = sWT + nC * KK;

    v8f acc = {0.f, 0.f, 0.f, 0.f, 0.f, 0.f, 0.f, 0.f};
#pragma unroll
    for (int k0 = 0; k0 < KK; k0 += 4) {
        uint32_t ka = k0 + 2 * half;
        v2f a  = *(const v2f*)(Ar + ka);   // A[m][ka..ka+1], 8B aligned
        v2f bb = *(const v2f*)(Bc + ka);   // W[ka..ka+1][n], 8B aligned (LDS)
        acc = __builtin_amdgcn_wmma_f32_16x16x4_f32(false, a, false, bb,
                                                    (short)0, acc, false, false);
    }
    float bn = bias.p[s] ? bias.p[s][nC] : 0.f;
    uint32_t m0   = rowBase + tm * 16 + 8 * half;       // row of acc[0]
    uint32_t cOff = m0 * (uint32_t)ldc + (uint32_t)coff + nC;
    if (resid) {
        uint32_t rOff = m0 * (uint32_t)ldr + (uint32_t)coff + nC;
#pragma unroll
        for (int r = 0; r < 8; ++r)
            C[cOff + (uint32_t)r * ldc] = acc[r] + bn + resid[rOff + (uint32_t)r * ldr];
    } else {
#pragma unroll
        for (int r = 0; r < 8; ++r)
            C[cOff + (uint32_t)r * ldc] = acc[r] + bn;
    }
}

// ---------------- causal depthwise conv (K=4) + SiLU: xc half of xz -> XC ----------------
__global__ void k_conv_silu(const float* __restrict__ XZ, Ptr4 cw, Ptr4 cb, float* __restrict__ XC)
{
    uint32_t t = blockIdx.x * blockDim.x + threadIdx.x;
    if (t >= (uint32_t)NS * B_N * L_N * DI) return;
    uint32_t e = t & 63;
    uint32_t r = t >> 6;               // row in (NS*B*L)
    uint32_t l = r & (L_N - 1);
    uint32_t s = r >> 17;
    const float* w = cw.p[s] + e * 4;
    float acc = cb.p[s][e];
    uint32_t rowBase = r - l;
#pragma unroll
    for (int k = 0; k < 4; ++k) {
        int lt = (int)l + k - 3;
        if (lt >= 0) acc += XZ[(rowBase + (uint32_t)lt) * (2 * DI) + e] * w[k];
    }
    XC[r * DI + e] = siluf_(acc);
}

// ---------------- selective scan: thread = (stream,b,channel), h[16]/A[16] in VGPRs ------
__global__ void k_scan(const float* __restrict__ XC, const float* __restrict__ DT,
                       const float* __restrict__ PR, const float* __restrict__ XZ,
                       Ptr4 Alog, Ptr4 bdt, Ptr4 Dp, float* __restrict__ Y)
{
    uint32_t t = blockIdx.x * blockDim.x + threadIdx.x;
    if (t >= NS * B_N * DI) return;
    uint32_t e  = t & 63;
    uint32_t sb = t >> 6;        // s*B + b
    uint32_t s  = t >> 12;
    const float* al = Alog.p[s] + e * NSTATE;
    float Aa[NSTATE];
#pragma unroll
    for (int n = 0; n < NSTATE; ++n) Aa[n] = -__expf(al[n]);
    float bd = bdt.p[s][e];
    float Dv = Dp.p[s][e];
    float h[NSTATE];
#pragma unroll
    for (int n = 0; n < NSTATE; ++n) h[n] = 0.f;

    uint32_t base = sb * L_N;
    for (uint32_t l = 0; l < L_N; ++l) {
        uint32_t r = base + l;
        const float* pr = PR + r * PRW;
        __builtin_prefetch(pr + PRW, 0, 0);       // next timestep's B/C row
        float u  = XC[r * DI + e];
        float dr = DT[r * DI + e] + bd;
        float dt = (dr > 20.f) ? dr : log1pf(__expf(dr));  // softplus
        float du = dt * u;
        float y = 0.f;
        const v2f* prB = (const v2f*)(pr + 2);    // B = proj[...,2:18]  (8B aligned)
        const v2f* prC = (const v2f*)(pr + 18);   // C = proj[...,18:34] (8B aligned)
#pragma unroll
        for (int n2 = 0; n2 < NSTATE / 2; ++n2) {
            v2f Bp = prB[n2];
            v2f Cp = prC[n2];
            int n0 = 2 * n2, n1 = 2 * n2 + 1;
            float dA0 = __expf(dt * Aa[n0]);
            float dA1 = __expf(dt * Aa[n1]);
            h[n0] = fmaf(dA0, h[n0], du * Bp.x);
            h[n1] = fmaf(dA1, h[n1], du * Bp.y);
            y = fmaf(h[n0], Cp.x, y);
            y = fmaf(h[n1], Cp.y, y);
        }
        float z = XZ[r * (2 * DI) + DI + e];
        Y[r * DI + e] = (y + u * Dv) * siluf_(z);
    }
}

// ---------------- concat fwd||flip(bwd) for both streams ----------------
__global__ void k_concat(const float* __restrict__ X, float* __restrict__ CL, float* __restrict__ CS)
{
    uint32_t t = blockIdx.x * blockDim.x + threadIdx.x;
    if (t >= B_N * L_N * DM) return;
    uint32_t d = t & 31;
    uint32_t l = (t >> 5) & (L_N - 1);
    uint32_t b = t >> 16;
    uint32_t rf = b * L_N + l;
    uint32_t rb = b * L_N + (L_N - 1 - l);
    const uint32_t SB = B_N * L_N;
    CL[rf * (2 * DM) + d]      = X[(0 * SB + rf) * DM + d];
    CL[rf * (2 * DM) + DM + d] = X[(1 * SB + rb) * DM + d];
    CS[rf * (2 * DM) + d]      = X[(2 * SB + rf) * DM + d];
    CS[rf * (2 * DM) + DM + d] = X[(3 * SB + rb) * DM + d];
}

// ---------------- fusion gate elementwise: G = COMB * sigmoid(G) ----------------
__global__ void k_gate(const float* __restrict__ CB, float* __restrict__ G)
{
    uint32_t t = blockIdx.x * blockDim.x + threadIdx.x;
    if (t >= B_N * L_N * 64) return;
    G[t] = CB[t] * sigmoidf_(G[t]);
}

// ---------------- mean pool over L ----------------
__global__ void k_pool(const float* __restrict__ FU, float* __restrict__ P)
{
    __shared__ float red[256];
    uint32_t b = blockIdx.x;
    uint32_t t = threadIdx.x;
    uint32_t d = t & 63, c = t >> 6;      // 4 chunks of 512
    float sum = 0.f;
    for (uint32_t l = c * 512; l < (c + 1) * 512; ++l)
        sum += FU[(b * L_N + l) * 64 + d];
    red[t] = sum;
    __syncthreads();
    if (t < 64)
        P[b * 64 + d] = (red[t] + red[t + 64] + red[t + 128] + red[t + 192]) * (1.f / L_N);
}

// ---------------- host side ----------------
static inline Ptr4 same4(const float* p) { Ptr4 r; r.p[0] = r.p[1] = r.p[2] = r.p[3] = p; return r; }

extern "C" void kernel_launch(void* const* d_in, const int* in_sizes, int n_in,
                              void* d_out, int out_size, void* d_ws, size_t ws_size,
                              hipStream_t stream)
{
    (void)out_size; (void)ws_size;
    // Logical indices follow setup_inputs() insertion order; fallback table handles
    // jax-pytree sorted-key flattening (detected via in_sizes[0]).
    static const int MAP_SORTED[68] = {
        67,                         // x
        17, 18, 20, 19,             // len_embed: W,b,ln_s,ln_b
        49, 50, 52, 51,             // struct_cont
        0,                          // dir_emb
        32, 31, 24, 30, 29, 26, 23, 27, 21, 22, 25, 28,   // len_fwd   (ln_s,ln_b,W_in,conv_w,conv_b,W_xproj,W_dt,b_dt,A_log,D,W_out,b_out)
        16, 15,  8, 14, 13, 10,  7, 11,  5,  6,  9, 12,   // len_bwd
        64, 63, 56, 62, 61, 58, 55, 59, 53, 54, 57, 60,   // struct_fwd
        48, 47, 40, 46, 45, 42, 39, 43, 37, 38, 41, 44,   // struct_bwd
        33, 34, 65, 66, 1, 2, 3, 4, 35, 36                // len_proj, struct_proj, fusion_gate, fusion_proj, projection
    };
    const bool ins = (n_in > 0 && in_sizes && in_sizes[0] == B_N * L_N * 3);
    auto F = [&](int lg) -> const float* {
        return (const float*)d_in[ins ? lg : MAP_SORTED[lg]];
    };

    const size_t BL  = (size_t)B_N * L_N;   // 131072
    const size_t SBL = (size_t)NS * BL;

    float* ws  = (float*)d_ws;
    float* X   = ws;                        // (NS,B,L,32)
    float* XZ  = X   + SBL * DM;            // (NS,B,L,128)
    float* XC  = XZ  + SBL * (2 * DI);      // (NS,B,L,64)
    float* PR  = XC  + SBL * DI;            // (NS,B,L,48)
    float* DTb = PR  + SBL * PRW;           // (NS,B,L,64)   (H aliases first half)
    float* Yb  = DTb + SBL * DI;            // (NS,B,L,64)
    float* Hb  = DTb;
    // post-stack buffers alias XZ (dead after the stacks)
    float* CL = XZ;
    float* CS = XZ + BL * 64;
    float* CB = XZ + 2 * BL * 64;
    float* GB = XZ + 3 * BL * 64;
    float* FU = XZ + 4 * BL * 64;
    float* PO = XZ + 5 * BL * 64;

    auto gemm = [&](const float* A, int lda, Ptr4 W, int Kw, int Nw, Ptr4 bias,
                    const float* resid, int ldr, float* C, int ldc, int coff,
                    int Mrows, int K, int N, int nstream) {
        int tiles = (Mrows / 16) * (N / 16);
        dim3 g((tiles + 7) / 8, nstream);
        if (K == 4)
            k_gemm_wmma<4><<<g, 256, 0, stream>>>(A, lda, W, Kw, Nw, bias, resid, ldr,
                                                  C, ldc, coff, Mrows, N);
        else if (K == 32)
            k_gemm_wmma<32><<<g, 256, 0, stream>>>(A, lda, W, Kw, Nw, bias, resid, ldr,
                                                   C, ldc, coff, Mrows, N);
        else
            k_gemm_wmma<64><<<g, 256, 0, stream>>>(A, lda, W, Kw, Nw, bias, resid, ldr,
                                                   C, ldc, coff, Mrows, N);
    };
    Ptr4 nul{}; nul.p[0] = nul.p[1] = nul.p[2] = nul.p[3] = nullptr;

    // embed -> 4 stream inputs
    k_embed<<<(int)((BL + 255) / 256), 256, 0, stream>>>(
        F(0), F(1), F(2), F(3), F(4), F(5), F(6), F(7), F(8), F(9), X);

    const int SBASE[4] = {10, 22, 34, 46}; // len_fwd, len_bwd, struct_fwd, struct_bwd
    for (int layer = 0; layer < 2; ++layer) {
        Ptr4 lns, lnb, win, cw, cb4, wxp, wdt, bdt, alog, dvec, wout, bout;
        for (int s = 0; s < 4; ++s) {
            int b0 = SBASE[s];
            lns.p[s]  = F(b0 + 0) + layer * DM;
            lnb.p[s]  = F(b0 + 1) + layer * DM;
            win.p[s]  = F(b0 + 2) + layer * DM * (2 * DI);
            cw.p[s]   = F(b0 + 3) + layer * DI * 4;
            cb4.p[s]  = F(b0 + 4) + layer * DI;
            wxp.p[s]  = F(b0 + 5) + layer * DI * 34;
            wdt.p[s]  = F(b0 + 6) + layer * 2 * DI;
            bdt.p[s]  = F(b0 + 7) + layer * DI;
            alog.p[s] = F(b0 + 8) + layer * DI * NSTATE;
            dvec.p[s] = F(b0 + 9) + layer * DI;
            wout.p[s] = F(b0 + 10) + layer * DI * DM;
            bout.p[s] = F(b0 + 11) + layer * DM;
        }
        // 1) layernorm
        k_ln<<<(int)((SBL + 255) / 256), 256, 0, stream>>>(X, lns, lnb, Hb);
        // 2) xz = h @ W_in (32 -> 128)
        gemm(Hb, DM, win, DM, 2 * DI, nul, nullptr, 0, XZ, 2 * DI, 0, (int)BL, DM, 2 * DI, NS);
        // 3) xc = silu(causal_conv(xz[:, :, :64]))
        k_conv_silu<<<(int)((SBL * DI + 255) / 256), 256, 0, stream>>>(XZ, cw, cb4, XC);
        // 4) proj = xc @ W_xproj (64 -> 34, padded N=48)
        gemm(XC, DI, wxp, DI, 34, nul, nullptr, 0, PR, PRW, 0, (int)BL, DI, PRW, NS);
        // 5) dt_raw = proj[:, :, :2] @ W_dt (K padded 2 -> 4; zero LDS rows kill cols 2,3)
        gemm(PR, PRW, wdt, 2, DI, nul, nullptr, 0, DTb, DI, 0, (int)BL, 4, DI, NS);
        // 6) selective scan + D skip + z gating
        k_scan<<<(NS * B_N * DI + 255) / 256, 256, 0, stream>>>(XC, DTb, PR, XZ, alog, bdt, dvec, Yb);
        // 7) x = x + y @ W_out + b_out  (in-place residual: each element RAW within one lane)
        gemm(Yb, DI, wout, DI, DM, bout, X, DM, X, DM, 0, (int)BL, DI, DM, NS);
    }

    // concat fwd || unflipped-bwd for both feature groups
    k_concat<<<(int)((BL * DM + 255) / 256), 256, 0, stream>>>(X, CL, CS);
    // h_len / h_struct projections written into the two halves of CB (via coff)
    gemm(CL, 64, same4(F(58)), 64, DM, same4(F(59)), nullptr, 0, CB, 64, 0,  (int)BL, 64, DM, 1);
    gemm(CS, 64, same4(F(60)), 64, DM, same4(F(61)), nullptr, 0, CB, 64, DM, (int)BL, 64, DM, 1);
    // fusion gate: G = CB @ Wg + bg; then G = CB * sigmoid(G)
    gemm(CB, 64, same4(F(62)), 64, 64, same4(F(63)), nullptr, 0, GB, 64, 0, (int)BL, 64, 64, 1);
    k_gate<<<(int)((BL * 64 + 255) / 256), 256, 0, stream>>>(CB, GB);
    // fused = G @ Wp + bp
    gemm(GB, 64, same4(F(64)), 64, 64, same4(F(65)), nullptr, 0, FU, 64, 0, (int)BL, 64, 64, 1);
    // mean pool over L
    k_pool<<<B_N, 256, 0, stream>>>(FU, PO);
    // final projection (64,64) @ (64,128) + b -> d_out
    gemm(PO, 64, same4(F(66)), 64, 128, same4(F(67)), nullptr, 0, (float*)d_out, 128, 0, B_N, 64, 128, 1);
}